// WeaveConcat_33904471834982
// MI455X (gfx1250) — compile-verified
//
#include <hip/hip_runtime.h>
#include <math.h>

typedef __attribute__((ext_vector_type(2))) float v2f;
typedef __attribute__((ext_vector_type(4))) float v4f;
typedef __attribute__((ext_vector_type(8))) float v8f;

#define N_ROWS_C (8192 * 128)   // 1048576
#define FDIM 128
#define CHUNK 1024
#define NCHUNK (N_ROWS_C / CHUNK)   // 1024
#define TILES_PER_BLOCK 16
#define ROW_TILES (N_ROWS_C / 16)   // 65536

// ---------- Pass 1: per-chunk valid counts ----------
__global__ __launch_bounds__(256) void count_kernel(const int* __restrict__ mask,
                                                    int* __restrict__ counts) {
    __shared__ int sdata[256];
    const int base = blockIdx.x * CHUNK;
    const int t = threadIdx.x;
    int s = 0;
#pragma unroll
    for (int i = 0; i < 4; ++i) s += (mask[base + t * 4 + i] != 0) ? 1 : 0;
    sdata[t] = s;
    __syncthreads();
    for (int off = 128; off > 0; off >>= 1) {
        if (t < off) sdata[t] += sdata[t + off];
        __syncthreads();
    }
    if (t == 0) counts[blockIdx.x] = sdata[0];
}

// ---------- Pass 2: exclusive scan of the 1024 chunk counts ----------
__global__ __launch_bounds__(1024) void scan_kernel(const int* __restrict__ counts,
                                                    int* __restrict__ offsets,
                                                    int* __restrict__ nvalid) {
    __shared__ int sdata[NCHUNK];
    const int t = threadIdx.x;
    const int v = counts[t];
    sdata[t] = v;
    __syncthreads();
    for (int off = 1; off < NCHUNK; off <<= 1) {
        int x = (t >= off) ? sdata[t - off] : 0;
        __syncthreads();
        sdata[t] += x;
        __syncthreads();
    }
    offsets[t] = sdata[t] - v;           // exclusive prefix
    if (t == NCHUNK - 1) *nvalid = sdata[t];
}

// ---------- Pass 3: emit stable permutation order[dst] = src ----------
__global__ __launch_bounds__(256) void order_kernel(const int* __restrict__ mask,
                                                    const int* __restrict__ offsets,
                                                    const int* __restrict__ nvalidp,
                                                    int* __restrict__ order) {
    __shared__ int sdata[256];
    const int base = blockIdx.x * CHUNK;
    const int t = threadIdx.x;
    int m[4];
    int s = 0;
#pragma unroll
    for (int i = 0; i < 4; ++i) {
        m[i] = (mask[base + t * 4 + i] != 0) ? 1 : 0;
        s += m[i];
    }
    sdata[t] = s;
    __syncthreads();
    for (int off = 1; off < 256; off <<= 1) {
        int x = (t >= off) ? sdata[t - off] : 0;
        __syncthreads();
        sdata[t] += x;
        __syncthreads();
    }
    int pre = sdata[t] - s + offsets[blockIdx.x];  // global # valid before element t*4
    const int nv = *nvalidp;
#pragma unroll
    for (int i = 0; i < 4; ++i) {
        const int gi = base + t * 4 + i;
        const int dst = m[i] ? pre : (nv + (gi - pre));
        order[dst] = gi;
        pre += m[i];
    }
}

// ---------- Pass 4: gathered GEMM + bias + tanh via V_WMMA_F32_16X16X4_F32 ----------
// 256 threads = 8 waves; wave w owns output columns [16w, 16w+16).
// Double-buffered LDS staging with the barrier at the END of each iteration:
//   issue gather loads(t+1) -> WMMA compute(t) -> wait loads, ds_store alt buf -> barrier
// so the HBM gather latency is covered by the 32-WMMA chain within each wave.
// Non-temporal hints on the once-touched streams (features in, output out)
// keep W / bias / order hot in the 192 MB L2.
__global__ __launch_bounds__(256) void gemm_kernel(const float* __restrict__ feat,
                                                   const float* __restrict__ W,
                                                   const float* __restrict__ bias,
                                                   const int* __restrict__ order,
                                                   float* __restrict__ out) {
    __shared__ v4f sA[2][16 * 32];   // 2 x 8 KB: 16 rows x 128 f32 each

    const int t     = threadIdx.x;
    const int lane  = t & 31;
    const int wave  = t >> 5;          // 0..7
    const int laneM = lane & 15;       // M (A) / N (B,C,D) sub-index
    const int hi    = lane >> 4;       // selects K pair {0,1} vs {2,3}
    const int colBase = wave * 16;

    // Staging geometry: thread t handles float4 slots t and t+256 (512 total).
    const int row0 = t >> 5;              // 0..7
    const int c40  = t & 31;
    const int row1 = row0 + 8;            // 8..15

    // B fragments for full K=128: 32 k-steps x 2 VGPRs, loaded once per block.
    v2f Bf[32];
#pragma unroll
    for (int k = 0; k < 32; ++k) {
        const int kk = k * 4 + hi * 2;
        Bf[k].x = W[(kk + 0) * FDIM + colBase + laneM];
        Bf[k].y = W[(kk + 1) * FDIM + colBase + laneM];
    }
    const float bv = bias[colBase + laneM];

    const int tile0 = blockIdx.x * TILES_PER_BLOCK;

    // Prologue: stage tile 0 into buffer 0, then barrier.
    {
        const int tb = tile0 * 16;
        const int s0 = order[tb + row0];
        const int s1 = order[tb + row1];
        v4f p0 = __builtin_nontemporal_load((const v4f*)(feat + (size_t)s0 * FDIM) + c40);
        v4f p1 = __builtin_nontemporal_load((const v4f*)(feat + (size_t)s1 * FDIM) + c40);
        sA[0][t]       = p0;
        sA[0][t + 256] = p1;
    }
    __syncthreads();

    for (int tt = 0; tt < TILES_PER_BLOCK; ++tt) {
        const int tileBase = (tile0 + tt) * 16;
        const bool more = (tt + 1) < TILES_PER_BLOCK;

        // Issue next tile's gather loads first (consumed only after compute).
        v4f r0, r1;
        if (more) {
            const int tb = (tile0 + tt + 1) * 16;
            const int s0 = order[tb + row0];
            const int s1 = order[tb + row1];
            r0 = __builtin_nontemporal_load((const v4f*)(feat + (size_t)s0 * FDIM) + c40);
            r1 = __builtin_nontemporal_load((const v4f*)(feat + (size_t)s1 * FDIM) + c40);
        }

        // Compute current tile from buf[tt&1] (ready since last barrier).
        const float* cur = (const float*)sA[tt & 1];
        v8f acc = {};
#pragma unroll
        for (int k = 0; k < 32; ++k) {
            const int kk = k * 4 + hi * 2;
            v2f a;
            a.x = cur[laneM * FDIM + kk];
            a.y = cur[laneM * FDIM + kk + 1];
            acc = __builtin_amdgcn_wmma_f32_16x16x4_f32(
                /*neg_a=*/false, a, /*neg_b=*/false, Bf[k],
                /*c_mod=*/(short)0, acc, /*reuse_a=*/false, /*reuse_b=*/false);
        }

        // Park next tile in the alternate buffer (its previous readers all
        // finished before the barrier that ended iteration tt-1).
        if (more) {
            sA[(tt + 1) & 1][t]       = r0;
            sA[(tt + 1) & 1][t + 256] = r1;
        }

        // C/D layout: VGPR r -> M=r (lanes 0-15) / M=r+8 (lanes 16-31), N=laneM.
        float* ocol = out + (size_t)tileBase * FDIM + colBase + laneM;
#pragma unroll
        for (int r = 0; r < 8; ++r) {
            const int row = r + hi * 8;
            __builtin_nontemporal_store(tanhf(acc[r] + bv), ocol + (size_t)row * FDIM);
        }

        // End-of-iteration barrier: publishes buf[(tt+1)&1] stores and closes
        // out all reads of buf[tt&1] before iteration tt+1 overwrites it.
        __syncthreads();
    }
}

extern "C" void kernel_launch(void* const* d_in, const int* in_sizes, int n_in,
                              void* d_out, int out_size, void* d_ws, size_t ws_size,
                              hipStream_t stream) {
    const float* feat = (const float*)d_in[0];   // [1048576, 128] f32
    const int*   mask = (const int*)d_in[1];     // [1048576] i32
    const float* W    = (const float*)d_in[2];   // [128, 128] f32
    const float* bias = (const float*)d_in[3];   // [128] f32
    float* out = (float*)d_out;                  // [1048576, 128] f32

    int* wsp      = (int*)d_ws;
    int* counts   = wsp;             // 1024 ints
    int* offsets  = wsp + 1024;      // 1024 ints
    int* nvalid   = wsp + 2048;      // 1 int
    int* order    = wsp + 2560;      // 1048576 ints (256B-aligned offset)

    count_kernel<<<NCHUNK, 256, 0, stream>>>(mask, counts);
    scan_kernel<<<1, 1024, 0, stream>>>(counts, offsets, nvalid);
    order_kernel<<<NCHUNK, 256, 0, stream>>>(mask, offsets, nvalid, order);
    gemm_kernel<<<ROW_TILES / TILES_PER_BLOCK, 256, 0, stream>>>(feat, W, bias, order, out);
}